// FSDLikeLoss_53953379172939
// MI455X (gfx1250) — compile-verified
//
#include <hip/hip_runtime.h>
#include <math.h>
#include <stdint.h>

// Problem shape (fixed by reference setup_inputs)
#define B_   8
#define C_   256
#define H_   112
#define W_   112
#define HW_  (H_ * W_)          // 12544
#define HW4_ (HW_ / 4)          // 3136
#define NC_  (B_ * C_)          // 2048
#define NHW_ ((float)(B_ * HW_))// 100352

#define USE_ASYNC_LDS 1

typedef __attribute__((ext_vector_type(2))) float f32x2;
typedef __attribute__((ext_vector_type(8))) float f32x8;

#if defined(__has_builtin)
#  if __has_builtin(__builtin_amdgcn_wmma_f32_16x16x4_f32)
#    define HAVE_WMMA_F32X4 1
#  endif
#endif

// ---------------------------------------------------------------------------
// Wave-level sum on the WMMA unit: A = per-lane (x, 0) forms a 16x4 f32 matrix
// (lane L -> A[L][0..1], lane L+16 -> A[L][2..3]); B = ones(4x16) makes every
// column of D the K-reduction. Each lane sums its 8 accumulator VGPRs (rows
// 0-7 on lanes 0-15, rows 8-15 on lanes 16-31); one shfl_xor(16) finishes the
// 32-lane reduction. EXEC must be all ones: only called at uniform points.
// ---------------------------------------------------------------------------
__device__ __forceinline__ float wave_sum(float x) {
#if defined(HAVE_WMMA_F32X4)
  f32x2 a; a.x = x; a.y = 0.0f;
  f32x2 ones; ones.x = 1.0f; ones.y = 1.0f;
  f32x8 c = {};
  c = __builtin_amdgcn_wmma_f32_16x16x4_f32(
      /*neg_a=*/false, a, /*neg_b=*/false, ones,
      /*c_mod=*/(short)0, c, /*reuse_a=*/false, /*reuse_b=*/false);
  float s = c[0] + c[1] + c[2] + c[3] + c[4] + c[5] + c[6] + c[7];
  s += __shfl_xor(s, 16, 32);
  return s;
#else
  for (int o = 16; o; o >>= 1) x += __shfl_xor(x, o, 32);
  return x;
#endif
}

__device__ __forceinline__ float wave_max(float x) {
  for (int o = 16; o; o >>= 1) x = fmaxf(x, __shfl_xor(x, o, 32));
  return x;
}

// Block reductions (blockDim.x == 256 -> 8 wave32 waves). Deterministic.
__device__ __forceinline__ float block_sum(float x, float* red) {
  float ws = wave_sum(x);
  int lane = threadIdx.x & 31, wv = threadIdx.x >> 5, nw = blockDim.x >> 5;
  __syncthreads();
  if (lane == 0) red[wv] = ws;
  __syncthreads();
  float t = red[0];
  for (int i = 1; i < nw; ++i) t += red[i];
  return t;
}

__device__ __forceinline__ float block_max(float x, float* red) {
  float wm = wave_max(x);
  int lane = threadIdx.x & 31, wv = threadIdx.x >> 5, nw = blockDim.x >> 5;
  __syncthreads();
  if (lane == 0) red[wv] = wm;
  __syncthreads();
  float t = red[0];
  for (int i = 1; i < nw; ++i) t = fmaxf(t, red[i]);
  return t;
}

// ---------------------------------------------------------------------------
// K1: per-batch Gaussian fg/bg weights from the mask. Writes sqrt(fg), sqrt(bg).
// Tiny (8 * 12544) -- scalar loops are fine.
// ---------------------------------------------------------------------------
__global__ void k_gauss(const float* __restrict__ fg,
                        float* __restrict__ sqfg, float* __restrict__ sqbg) {
  __shared__ float red[16];
  const int b = blockIdx.x;
  const float* f = fg + (size_t)b * HW_;

  float c0 = 0.f, sy = 0.f, sx = 0.f;
  for (int i = threadIdx.x; i < HW_; i += blockDim.x) {
    float m = f[i] > 0.5f ? 1.f : 0.f;
    float y = (float)(i / W_), x = (float)(i % W_);
    c0 += m; sy += m * y; sx += m * x;
  }
  float cnt  = block_sum(c0, red);
  float syT  = block_sum(sy, red);
  float sxT  = block_sum(sx, red);
  float safe = fmaxf(cnt, 1.f);
  float muy = syT / safe, mux = sxT / safe;

  float vy = 0.f, vx = 0.f;
  for (int i = threadIdx.x; i < HW_; i += blockDim.x) {
    float m = f[i] > 0.5f ? 1.f : 0.f;
    float y = (float)(i / W_), x = (float)(i % W_);
    float dy = y - muy, dx = x - mux;
    vy += m * dy * dy; vx += m * dx * dx;
  }
  float sigy = fmaxf(sqrtf(block_sum(vy, red) / safe), 1.f);
  float sigx = fmaxf(sqrtf(block_sum(vx, red) / safe), 1.f);
  if (cnt < 1.f) {  // uniform branch: cnt identical across block
    muy = (H_ - 1) * 0.5f; mux = (W_ - 1) * 0.5f;
    sigy = fmaxf(1.f, H_ / 4.f); sigx = fmaxf(1.f, W_ / 4.f);
  }
  float isy = 1.f / ((sigy + 1e-8f) * (sigy + 1e-8f));
  float isx = 1.f / ((sigx + 1e-8f) * (sigx + 1e-8f));

  float gs = 0.f;
  for (int i = threadIdx.x; i < HW_; i += blockDim.x) {
    float y = (float)(i / W_), x = (float)(i % W_);
    float dy = y - muy, dx = x - mux;
    gs += expf(-0.5f * (dy * dy * isy + dx * dx * isx));
  }
  float gsum = block_sum(gs, red);
  float ginv = gsum > 0.f ? 1.f / gsum : 1.f;
  float rs = 1.f / safe;

  float bs0 = 0.f;
  for (int i = threadIdx.x; i < HW_; i += blockDim.x) {
    float m = f[i] > 0.5f ? 1.f : 0.f;
    float y = (float)(i / W_), x = (float)(i % W_);
    float dy = y - muy, dx = x - mux;
    float g = expf(-0.5f * (dy * dy * isy + dx * dx * isx)) * ginv;
    float fused = fmaxf(m * rs, g * rs);
    sqfg[(size_t)b * HW_ + i] = sqrtf(fused);
    bs0 += (fused > 0.f) ? 0.f : 1.f;
  }
  float binv = 1.f / fmaxf(block_sum(bs0, red), 1.f);
  for (int i = threadIdx.x; i < HW_; i += blockDim.x) {
    float bm = (sqfg[(size_t)b * HW_ + i] > 0.f) ? 0.f : 1.f;
    sqbg[(size_t)b * HW_ + i] = sqrtf(bm * binv);
  }
}

// ---------------------------------------------------------------------------
// K2a: per-(b,c) spatial sum of |teacher|. One block per row; b128 loads.
// ---------------------------------------------------------------------------
__global__ void k_csum(const float* __restrict__ t, float* __restrict__ csum) {
  __shared__ float red[16];
  const size_t r = blockIdx.x;
  const float4* p = reinterpret_cast<const float4*>(t + r * (size_t)HW_);
  float a = 0.f;
  for (int i = threadIdx.x; i < HW4_; i += blockDim.x) {
    float4 v = p[i];
    a += fabsf(v.x) + fabsf(v.y) + fabsf(v.z) + fabsf(v.w);
  }
  float s = block_sum(a, red);
  if (threadIdx.x == 0) csum[r] = s;
}

// K2b: per-(b,hw) channel sum of |teacher| (teacher L2-resident after K2a).
// Each thread owns 4 consecutive hw positions -> b128 loads/stores.
__global__ void k_chsum(const float* __restrict__ t, float* __restrict__ chsum) {
  const int b = blockIdx.y;
  const int i4 = blockIdx.x * blockDim.x + threadIdx.x;   // float4 index in HW
  if (i4 >= HW4_) return;
  const float4* p = reinterpret_cast<const float4*>(t + ((size_t)b * C_) * HW_) + i4;
  float4 a = make_float4(0.f, 0.f, 0.f, 0.f);
  for (int c = 0; c < C_; ++c) {
    float4 v = p[(size_t)c * HW4_];
    a.x += fabsf(v.x); a.y += fabsf(v.y); a.z += fabsf(v.z); a.w += fabsf(v.w);
  }
  reinterpret_cast<float4*>(chsum + (size_t)b * HW_)[i4] = a;
}

// K3: spatial softmax -> sa = sqrt(HW * softmax(chsum/C)). One block / batch.
__global__ void k_sattn(const float* __restrict__ chsum, float* __restrict__ sa) {
  __shared__ float red[16];
  const int b = blockIdx.x;
  const float4* p = reinterpret_cast<const float4*>(chsum + (size_t)b * HW_);
  const float invC = 1.f / (float)C_;
  float mx = -INFINITY;
  for (int i = threadIdx.x; i < HW4_; i += blockDim.x) {
    float4 v = p[i];
    mx = fmaxf(mx, fmaxf(fmaxf(v.x, v.y), fmaxf(v.z, v.w)) * invC);
  }
  float M = block_max(mx, red);
  float e = 0.f;
  for (int i = threadIdx.x; i < HW4_; i += blockDim.x) {
    float4 v = p[i];
    e += expf(v.x * invC - M) + expf(v.y * invC - M) +
         expf(v.z * invC - M) + expf(v.w * invC - M);
  }
  float invS = 1.f / block_sum(e, red);
  float4* o = reinterpret_cast<float4*>(sa + (size_t)b * HW_);
  for (int i = threadIdx.x; i < HW4_; i += blockDim.x) {
    float4 v = p[i], w;
    w.x = sqrtf((float)HW_ * expf(v.x * invC - M) * invS);
    w.y = sqrtf((float)HW_ * expf(v.y * invC - M) * invS);
    w.z = sqrtf((float)HW_ * expf(v.z * invC - M) * invS);
    w.w = sqrtf((float)HW_ * expf(v.w * invC - M) * invS);
    o[i] = w;
  }
}

// K4: channel softmax -> sqc = sqrt(C * softmax(csum/HW)). blockDim == C_.
__global__ void k_cattn(const float* __restrict__ csum, float* __restrict__ sqc) {
  __shared__ float red[16];
  const int b = blockIdx.x;
  float v = csum[(size_t)b * C_ + threadIdx.x] * (1.f / (float)HW_);
  float M = block_max(v, red);
  float e = expf(v - M);
  float S = block_sum(e, red);
  sqc[(size_t)b * C_ + threadIdx.x] = sqrtf((float)C_ * e / S);
}

// K4b: fold sqrt(s_attn) into the fg/bg weights (shared across all channels).
__global__ void k_wmaps(const float* __restrict__ sa,
                        const float* __restrict__ sqfg, const float* __restrict__ sqbg,
                        float* __restrict__ wfg, float* __restrict__ wbg) {
  const size_t i = (size_t)blockIdx.x * blockDim.x + threadIdx.x; // float4 idx
  float4 s = reinterpret_cast<const float4*>(sa)[i];
  float4 f = reinterpret_cast<const float4*>(sqfg)[i];
  float4 g = reinterpret_cast<const float4*>(sqbg)[i];
  float4 of, og;
  of.x = s.x * f.x; of.y = s.y * f.y; of.z = s.z * f.z; of.w = s.w * f.w;
  og.x = s.x * g.x; og.y = s.y * g.y; og.z = s.z * g.z; og.w = s.w * g.w;
  reinterpret_cast<float4*>(wfg)[i] = of;
  reinterpret_cast<float4*>(wbg)[i] = og;
}

// ---------------------------------------------------------------------------
// K5: per-channel sum / sumsq over (b,hw) for the 4 weighted tensors.
// One block per channel; b128 loads, fully coalesced.
// ---------------------------------------------------------------------------
__global__ void k_stats(const float* __restrict__ t, const float* __restrict__ s,
                        const float* __restrict__ wfg, const float* __restrict__ wbg,
                        const float* __restrict__ sqc, float* __restrict__ stats) {
  __shared__ float red[16];
  const int c = blockIdx.x;
  float a0=0,a1=0,a2=0,a3=0,a4=0,a5=0,a6=0,a7=0;
  for (int b = 0; b < B_; ++b) {
    const float sc = sqc[(size_t)b * C_ + c];
    const float4* tp = reinterpret_cast<const float4*>(t + ((size_t)b * C_ + c) * HW_);
    const float4* sp = reinterpret_cast<const float4*>(s + ((size_t)b * C_ + c) * HW_);
    const float4* wf = reinterpret_cast<const float4*>(wfg + (size_t)b * HW_);
    const float4* wb = reinterpret_cast<const float4*>(wbg + (size_t)b * HW_);
    for (int i = threadIdx.x; i < HW4_; i += blockDim.x) {
      float4 tv = tp[i], sv = sp[i], fv = wf[i], gv = wb[i];
#define ACC_ONE(TT, SS, FF, GG)                                    \
      { float f = (FF) * sc, g = (GG) * sc, z;                     \
        z = (TT) * f; a0 += z; a1 += z * z;                        \
        z = (SS) * f; a2 += z; a3 += z * z;                        \
        z = (TT) * g; a4 += z; a5 += z * z;                        \
        z = (SS) * g; a6 += z; a7 += z * z; }
      ACC_ONE(tv.x, sv.x, fv.x, gv.x)
      ACC_ONE(tv.y, sv.y, fv.y, gv.y)
      ACC_ONE(tv.z, sv.z, fv.z, gv.z)
      ACC_ONE(tv.w, sv.w, fv.w, gv.w)
#undef ACC_ONE
    }
  }
  float acc[8] = {a0,a1,a2,a3,a4,a5,a6,a7};
  for (int k = 0; k < 8; ++k) {
    float tot = block_sum(acc[k], red);
    if (threadIdx.x == 0) stats[(size_t)c * 8 + k] = tot;
  }
}

// K5b: invsd[c*4+k] = 1/(std+1e-6), std with ddof=1 over N = B*HW.
__global__ void k_invsd(const float* __restrict__ stats, float* __restrict__ invsd) {
  const int idx = blockIdx.x * blockDim.x + threadIdx.x;
  if (idx >= C_ * 4) return;
  const int c = idx >> 2, k = idx & 3;
  float sum = stats[(size_t)c * 8 + k * 2];
  float ss  = stats[(size_t)c * 8 + k * 2 + 1];
  float mean = sum / NHW_;
  float var = (ss - NHW_ * mean * mean) / (NHW_ - 1.f);
  invsd[idx] = 1.f / (sqrtf(fmaxf(var, 0.f)) + 1e-6f);
}

// ---------------------------------------------------------------------------
// K6: per-(b,c) row KL. Stage t/s rows into LDS once with CDNA5 async copies
// (ASYNCcnt path, global_load_async_to_lds_b128), then 4 passes (max+sum x
// fg/bg) run from LDS with b128 ds loads. Weight rows come from L2 (only
// 800 KB total, fully resident). LDS = 100 KB -> multiple WGs per WGP.
// KL_row = (T1 - T2)/St - (Mt + log St) + (Ms + log Ss)
// (the mean subtraction of _feature_norm cancels inside log_softmax).
// ---------------------------------------------------------------------------
__global__ void k_kl(const float* __restrict__ t, const float* __restrict__ s,
                     const float* __restrict__ wfg, const float* __restrict__ wbg,
                     const float* __restrict__ sqc, const float* __restrict__ invsd,
                     float* __restrict__ klrows) {
  extern __shared__ float smem[];           // 2 * HW_ floats = 100,352 B
  __shared__ float red[16];
  float* tl = smem;
  float* sl = smem + HW_;
  const int r = blockIdx.x;
  const int b = r / C_, c = r % C_;
  const float* tp = t + (size_t)r * HW_;
  const float* sp = s + (size_t)r * HW_;

#if USE_ASYNC_LDS
  {
    uint32_t lt = (uint32_t)(uintptr_t)tl;
    uint32_t ls = (uint32_t)(uintptr_t)sl;
    uint64_t gt = (uint64_t)(uintptr_t)tp;
    uint64_t gs = (uint64_t)(uintptr_t)sp;
    for (int i = threadIdx.x; i < HW4_; i += blockDim.x) {
      uint32_t lo = (uint32_t)i << 4;       // 16B chunks
      asm volatile("global_load_async_to_lds_b128 %0, %1, off"
                   :: "v"(lt + lo), "v"(gt + (uint64_t)lo) : "memory");
      asm volatile("global_load_async_to_lds_b128 %0, %1, off"
                   :: "v"(ls + lo), "v"(gs + (uint64_t)lo) : "memory");
    }
    asm volatile("s_wait_asynccnt 0x0" ::: "memory");
  }
  __syncthreads();
#else
  {
    const float4* tp4 = reinterpret_cast<const float4*>(tp);
    const float4* sp4 = reinterpret_cast<const float4*>(sp);
    float4* tl4 = reinterpret_cast<float4*>(tl);
    float4* sl4 = reinterpret_cast<float4*>(sl);
    for (int i = threadIdx.x; i < HW4_; i += blockDim.x) {
      tl4[i] = tp4[i]; sl4[i] = sp4[i];
    }
  }
  __syncthreads();
#endif

  const float sc = sqc[(size_t)b * C_ + c];
  const float4* tl4 = reinterpret_cast<const float4*>(tl);
  const float4* sl4 = reinterpret_cast<const float4*>(sl);
  const float4* w0 = reinterpret_cast<const float4*>(wfg + (size_t)b * HW_);
  const float4* w1 = reinterpret_cast<const float4*>(wbg + (size_t)b * HW_);

  for (int pass = 0; pass < 2; ++pass) {
    const float4* w4 = pass == 0 ? w0 : w1;
    const float kt = sc * invsd[(size_t)c * 4 + pass * 2 + 0];
    const float ks = sc * invsd[(size_t)c * 4 + pass * 2 + 1];

    float mt = -INFINITY, ms = -INFINITY;
    for (int i = threadIdx.x; i < HW4_; i += blockDim.x) {
      float4 tv = tl4[i], sv = sl4[i], wv = w4[i];
      mt = fmaxf(mt, fmaxf(fmaxf(tv.x * wv.x, tv.y * wv.y),
                           fmaxf(tv.z * wv.z, tv.w * wv.w)) * kt);
      ms = fmaxf(ms, fmaxf(fmaxf(sv.x * wv.x, sv.y * wv.y),
                           fmaxf(sv.z * wv.z, sv.w * wv.w)) * ks);
    }
    float Mt = block_max(mt, red);
    float Ms = block_max(ms, red);

    float St = 0.f, T1 = 0.f, T2 = 0.f, Ss = 0.f;
    for (int i = threadIdx.x; i < HW4_; i += blockDim.x) {
      float4 tv = tl4[i], sv = sl4[i], wv = w4[i];
#define KL_ONE(TT, SS, WW)                                         \
      { float at = (TT) * (WW) * kt, av = (SS) * (WW) * ks;        \
        float et = expf(at - Mt);                                  \
        St += et; T1 += et * at; T2 += et * av;                    \
        Ss += expf(av - Ms); }
      KL_ONE(tv.x, sv.x, wv.x)
      KL_ONE(tv.y, sv.y, wv.y)
      KL_ONE(tv.z, sv.z, wv.z)
      KL_ONE(tv.w, sv.w, wv.w)
#undef KL_ONE
    }
    float StT = block_sum(St, red);
    float T1T = block_sum(T1, red);
    float T2T = block_sum(T2, red);
    float SsT = block_sum(Ss, red);
    if (threadIdx.x == 0) {
      float kl = (T1T - T2T) / StT - (Mt + logf(StT)) + (Ms + logf(SsT));
      klrows[(size_t)pass * NC_ + r] = kl;
    }
  }
}

// K7: final deterministic reduction -> scalar loss.
__global__ void k_final(const float* __restrict__ klrows, float* __restrict__ out) {
  __shared__ float red[16];
  float a = 0.f;
  for (int r = threadIdx.x; r < NC_; r += blockDim.x)
    a += 1.0f * klrows[r] + 0.2f * klrows[NC_ + r];
  float s = block_sum(a, red);
  if (threadIdx.x == 0) out[0] = s / (float)NC_;
}

// ---------------------------------------------------------------------------
extern "C" void kernel_launch(void* const* d_in, const int* in_sizes, int n_in,
                              void* d_out, int out_size, void* d_ws, size_t ws_size,
                              hipStream_t stream) {
  const float* s_feat = (const float*)d_in[0];
  const float* t_feat = (const float*)d_in[1];
  const float* fg_map = (const float*)d_in[2];
  // d_in[3] (bg_map) is unused: the reference overwrites it from fg_map.

  float* ws = (float*)d_ws;
  float* chsum = ws;                 // B*HW = 100352
  float* sa    = chsum + (size_t)B_ * HW_;
  float* sqfg  = sa    + (size_t)B_ * HW_;
  float* sqbg  = sqfg  + (size_t)B_ * HW_;
  float* wfg   = sqbg  + (size_t)B_ * HW_;
  float* wbg   = wfg   + (size_t)B_ * HW_;
  float* csum  = wbg   + (size_t)B_ * HW_;   // NC
  float* sqc   = csum  + NC_;                // NC
  float* stats = sqc   + NC_;                // C*8
  float* invsd = stats + (size_t)C_ * 8;     // C*4
  float* klrow = invsd + (size_t)C_ * 4;     // 2*NC

  const dim3 blk(256);
  k_gauss <<<B_, blk, 0, stream>>>(fg_map, sqfg, sqbg);
  k_csum  <<<NC_, blk, 0, stream>>>(t_feat, csum);
  k_chsum <<<dim3((HW4_ + 255) / 256, B_), blk, 0, stream>>>(t_feat, chsum);
  k_sattn <<<B_, blk, 0, stream>>>(chsum, sa);
  k_cattn <<<B_, blk, 0, stream>>>(csum, sqc);
  k_wmaps <<<(B_ * HW4_) / 256, blk, 0, stream>>>(sa, sqfg, sqbg, wfg, wbg);
  k_stats <<<C_, blk, 0, stream>>>(t_feat, s_feat, wfg, wbg, sqc, stats);
  k_invsd <<<4, blk, 0, stream>>>(stats, invsd);
  k_kl    <<<NC_, blk, 2 * HW_ * sizeof(float), stream>>>(t_feat, s_feat, wfg, wbg,
                                                          sqc, invsd, klrow);
  k_final <<<1, blk, 0, stream>>>(klrow, (float*)d_out);
}